// AdditiveAttention_15015205667427
// MI455X (gfx1250) — compile-verified
//
#include <hip/hip_runtime.h>
#include <hip/hip_bf16.h>
#include <math.h>

// Problem constants (match reference)
#define B_ 4
#define Q_ 512
#define K_ 512
#define D_ 256
#define H_ 256
#define V_ 256
#define HP_ (H_ + 4)      // padded LDS row: 260 floats -> row r starts at bank 4r
#define NEG_ (-1e6f)

typedef __attribute__((ext_vector_type(2))) float v2f;
typedef __attribute__((ext_vector_type(8))) float v8f;

// Native tanh when gfx1250 exposes it; exp2-based fallback otherwise.
__device__ __forceinline__ float fast_tanh(float x) {
#if __has_builtin(__builtin_amdgcn_tanhf)
  return __builtin_amdgcn_tanhf(x);
#elif __has_builtin(__builtin_amdgcn_exp2f)
  // tanh(x) = 1 - 2/(exp2(2*log2(e)*x) + 1)
  float e = __builtin_amdgcn_exp2f(x * 2.88539008177792681472f);
  return 1.0f - 2.0f / (e + 1.0f);
#else
  return tanhf(x);
#endif
}

__device__ __forceinline__ float fast_exp(float x) {
#if __has_builtin(__builtin_amdgcn_exp2f)
  return __builtin_amdgcn_exp2f(x * 1.44269504088896340736f);
#else
  return expf(x);
#endif
}

// CDNA5 async copy: global -> LDS, 16 bytes per lane, tracked by ASYNCcnt.
__device__ __forceinline__ void async_copy_b128(void* lds_dst, const void* gsrc) {
  unsigned lds_off = (unsigned)(size_t)lds_dst;            // LDS offset = low 32 bits of the pointer
  unsigned long long ga = (unsigned long long)(size_t)gsrc;
  asm volatile("global_load_async_to_lds_b128 %0, %1, off"
               :: "v"(lds_off), "v"(ga) : "memory");
}
__device__ __forceinline__ void async_wait0() {
  asm volatile("s_wait_asynccnt 0x0" ::: "memory");
}

// ---------------------------------------------------------------------------
// Kernel 1: Y[row, h] = sum_d X[row, d] * W[h, d]   (fp32, via WMMA 16x16x4)
// One wave (32 threads) per 16x16 output tile.
// ---------------------------------------------------------------------------
__global__ void k_project(const float* __restrict__ X, const float* __restrict__ W,
                          float* __restrict__ Y) {
  const int lane = threadIdx.x;          // 0..31
  const int m    = lane & 15;
  const int kk   = (lane >> 4) * 2;      // 0 or 2
  const int row0 = blockIdx.x * 16;
  const int h0   = blockIdx.y * 16;

  const float* __restrict__ arow = X + (size_t)(row0 + m) * D_;
  const float* __restrict__ brow = W + (size_t)(h0 + m) * D_;   // W is (H, D)

  v8f c = {};
#pragma unroll 8
  for (int d = 0; d < D_; d += 4) {
    v2f a, b;
    a.x = arow[d + kk];
    a.y = arow[d + kk + 1];
    b.x = brow[d + kk];
    b.y = brow[d + kk + 1];
    c = __builtin_amdgcn_wmma_f32_16x16x4_f32(false, a, false, b, (short)0, c,
                                              false, false);
  }

  const int rbase = row0 + ((lane >> 4) * 8);
#pragma unroll
  for (int i = 0; i < 8; ++i)
    Y[(size_t)(rbase + i) * H_ + h0 + m] = c[i];
}

// ---------------------------------------------------------------------------
// Kernel 2: scores[b,q,k] = sum_h w_v[h] * tanh(qp[b,q,h] + kp[b,k,h]), masked.
// 256-thread block computes a 16(q) x 16(k) tile.
//   - qp/kp tiles staged via global_load_async_to_lds_b128 (ASYNCcnt path)
//   - LDS rows padded to 260 floats -> conflict-free b128 reads across 16 rows
//   - inner loop: 3x ds_load_b128 per 4 h-steps, then add/tanh/fma (trans-bound)
// ---------------------------------------------------------------------------
__global__ void k_scores(const float* __restrict__ qp, const float* __restrict__ kp,
                         const float* __restrict__ wv, const int* __restrict__ vlen,
                         float* __restrict__ S) {
  __shared__ float qs[16][HP_];
  __shared__ float ks_[16][HP_];
  __shared__ float wvs[H_];

  const int b   = blockIdx.z;
  const int q0  = blockIdx.y * 16;
  const int k0  = blockIdx.x * 16;
  const int tid = threadIdx.x;           // 0..255

  wvs[tid] = wv[tid];

  // Stage 16x256 fp32 tiles of qp and kp: 1024 float4 each, 4 per thread.
#pragma unroll
  for (int i = 0; i < 4; ++i) {
    const int idx = tid + i * 256;       // float4 index within tile
    const int r   = idx >> 6;            // 64 float4 per row
    const int c4  = (idx & 63) * 4;      // float column
    async_copy_b128(&qs[r][c4],  qp + (size_t)(b * Q_ + q0 + r) * H_ + c4);
    async_copy_b128(&ks_[r][c4], kp + (size_t)(b * K_ + k0 + r) * H_ + c4);
  }
  async_wait0();          // this wave's async loads complete
  __syncthreads();        // all waves' tiles visible

  const int ql = tid >> 4;
  const int kl = tid & 15;

  const float4* __restrict__ qv = (const float4*)&qs[ql][0];
  const float4* __restrict__ kv = (const float4*)&ks_[kl][0];
  const float4* __restrict__ wq = (const float4*)&wvs[0];

  float s = 0.0f;
#pragma unroll 4
  for (int h4 = 0; h4 < H_ / 4; ++h4) {
    const float4 a = qv[h4];
    const float4 c = kv[h4];
    const float4 w = wq[h4];
    s = fmaf(w.x, fast_tanh(a.x + c.x), s);
    s = fmaf(w.y, fast_tanh(a.y + c.y), s);
    s = fmaf(w.z, fast_tanh(a.z + c.z), s);
    s = fmaf(w.w, fast_tanh(a.w + c.w), s);
  }

  const int kg = k0 + kl;
  if (kg >= vlen[b]) s = NEG_;
  S[(size_t)(b * Q_ + q0 + ql) * K_ + kg] = s;
}

// ---------------------------------------------------------------------------
// Kernel 3: in-place softmax over K=512 per (b,q) row. blockDim=256, 2 elem/thr.
// ---------------------------------------------------------------------------
__global__ void k_softmax(float* __restrict__ S) {
  float* __restrict__ p = S + (size_t)(blockIdx.y * Q_ + blockIdx.x) * K_;
  const int tid = threadIdx.x;           // 0..255

  __shared__ float red[256];

  const float v0 = p[tid];
  const float v1 = p[tid + 256];

  red[tid] = fmaxf(v0, v1);
  __syncthreads();
#pragma unroll
  for (int off = 128; off > 0; off >>= 1) {
    if (tid < off) red[tid] = fmaxf(red[tid], red[tid + off]);
    __syncthreads();
  }
  const float mx = red[0];
  __syncthreads();

  const float e0 = fast_exp(v0 - mx);
  const float e1 = fast_exp(v1 - mx);
  red[tid] = e0 + e1;
  __syncthreads();
#pragma unroll
  for (int off = 128; off > 0; off >>= 1) {
    if (tid < off) red[tid] = red[tid] + red[tid + off];
    __syncthreads();
  }
  const float inv = 1.0f / red[0];

  p[tid]       = e0 * inv;
  p[tid + 256] = e1 * inv;
}

// ---------------------------------------------------------------------------
// Kernel 4: out[b,q,v] = sum_k P[b,q,k] * Vals[b,k,v]  (fp32, WMMA 16x16x4)
// One wave per 16x16 output tile, K-loop over 512 in steps of 4.
// ---------------------------------------------------------------------------
__global__ void k_pv(const float* __restrict__ P, const float* __restrict__ Vm,
                     float* __restrict__ O) {
  const int lane = threadIdx.x;
  const int m    = lane & 15;
  const int kk   = (lane >> 4) * 2;
  const int b    = blockIdx.z;
  const int q0   = blockIdx.x * 16;
  const int v0   = blockIdx.y * 16;

  const float* __restrict__ arow = P + (size_t)(b * Q_ + q0 + m) * K_;
  const float* __restrict__ vb   = Vm + (size_t)b * K_ * V_;

  v8f c = {};
#pragma unroll 4
  for (int k = 0; k < K_; k += 4) {
    v2f a, bf;
    a.x  = arow[k + kk];
    a.y  = arow[k + kk + 1];
    bf.x = vb[(size_t)(k + kk) * V_ + v0 + m];
    bf.y = vb[(size_t)(k + kk + 1) * V_ + v0 + m];
    c = __builtin_amdgcn_wmma_f32_16x16x4_f32(false, a, false, bf, (short)0, c,
                                              false, false);
  }

  const int rbase = q0 + ((lane >> 4) * 8);
#pragma unroll
  for (int i = 0; i < 8; ++i)
    O[(size_t)(b * Q_ + rbase + i) * V_ + v0 + m] = c[i];
}

// ---------------------------------------------------------------------------
extern "C" void kernel_launch(void* const* d_in, const int* in_sizes, int n_in,
                              void* d_out, int out_size, void* d_ws, size_t ws_size,
                              hipStream_t stream) {
  (void)in_sizes; (void)n_in; (void)out_size; (void)ws_size;

  const float* queries = (const float*)d_in[0];   // (B,Q,D)
  const float* keys    = (const float*)d_in[1];   // (B,K,D)
  const float* values  = (const float*)d_in[2];   // (B,K,V)
  const int*   vlens   = (const int*)  d_in[3];   // (B,)
  const float* W_q     = (const float*)d_in[4];   // (H,D)
  const float* W_k     = (const float*)d_in[5];   // (H,D)
  const float* w_v     = (const float*)d_in[6];   // (H,)
  float*       out     = (float*)d_out;           // (B,Q,V)

  // Workspace layout (fp32): qp (B*Q*H) | kp (B*K*H) | scores (B*Q*K) = 8 MB
  float* qp = (float*)d_ws;
  float* kp = qp + (size_t)B_ * Q_ * H_;
  float* S  = kp + (size_t)B_ * K_ * H_;

  // 1) Projections via fp32 WMMA
  k_project<<<dim3((B_ * Q_) / 16, H_ / 16), 32, 0, stream>>>(queries, W_q, qp);
  k_project<<<dim3((B_ * K_) / 16, H_ / 16), 32, 0, stream>>>(keys,    W_k, kp);

  // 2) tanh feature-reduce + mask (trans/VALU bound: 268M v_tanh)
  k_scores<<<dim3(K_ / 16, Q_ / 16, B_), 256, 0, stream>>>(qp, kp, w_v, vlens, S);

  // 3) Row softmax over K
  k_softmax<<<dim3(Q_, B_), 256, 0, stream>>>(S);

  // 4) P @ V via fp32 WMMA
  k_pv<<<dim3(Q_ / 16, V_ / 16, B_), 32, 0, stream>>>(S, values, out);
}